// EuclideanDistanceHashDecoder_46694884442217
// MI455X (gfx1250) — compile-verified
//
#include <hip/hip_runtime.h>
#include <math.h>

// Problem constants (fixed by the reference): z is [50000, 512] f32,
// edge_index is [2, 150000] i64, output is 150000 f32.
#define D_FEAT 512
#define EPSF   1e-6f

#define WPB           4                  // waves per block (wave32)
#define BLOCK_THREADS (WPB * 32)
#define MAX_BLOCKS    2048               // 8192 waves -> ~18 edges/wave pipeline depth
#define CHUNKS        (D_FEAT / 128)     // float4 chunks per lane per row (=4)

// ---- CDNA5 async global->LDS path (guarded; feedback loop verifies lowering) ----
#if defined(__AMDGCN__) && __has_builtin(__builtin_amdgcn_global_load_async_to_lds_b128)
  #define USE_ASYNC 1
#else
  #define USE_ASYNC 0
#endif

#if defined(__AMDGCN__) && __has_builtin(__builtin_amdgcn_s_wait_asynccnt)
  #define WAIT_ASYNC(n) do { __builtin_amdgcn_s_wait_asynccnt(n); asm volatile("" ::: "memory"); } while (0)
#elif defined(__AMDGCN__)
  #define WAIT_ASYNC(n) asm volatile("s_wait_asynccnt %0" :: "i"(n) : "memory")
#else
  #define WAIT_ASYNC(n) do { } while (0)
#endif

// The builtin takes pointers to int4 vectors: param0 = global (AS1) source,
// param1 = LDS (AS3) destination (clang prints AS1 as '__device__' in HIP).
typedef int v4i __attribute__((vector_size(16)));
typedef __attribute__((address_space(1))) v4i gv4i_t;  // global int4
typedef __attribute__((address_space(3))) v4i lv4i_t;  // LDS int4

// Butterfly sum across the 32 lanes of a wave; every lane gets the total.
__device__ __forceinline__ float wave_sum32(float v) {
#pragma unroll
  for (int m = 16; m > 0; m >>= 1) v += __shfl_xor(v, m, 32);
  return v;
}

// av/bv hold this lane's 16 elements of rows a and b.
// Returns sigmoid(1 - || a/|a| - b/|b| + eps ||) on every lane.
__device__ __forceinline__ float edge_value(const float4* av, const float4* bv) {
  float aa = 0.f, bb = 0.f;
#pragma unroll
  for (int k = 0; k < CHUNKS; ++k) {
    aa = fmaf(av[k].x, av[k].x, aa);
    aa = fmaf(av[k].y, av[k].y, aa);
    aa = fmaf(av[k].z, av[k].z, aa);
    aa = fmaf(av[k].w, av[k].w, aa);
    bb = fmaf(bv[k].x, bv[k].x, bb);
    bb = fmaf(bv[k].y, bv[k].y, bb);
    bb = fmaf(bv[k].z, bv[k].z, bb);
    bb = fmaf(bv[k].w, bv[k].w, bb);
  }
  aa = wave_sum32(aa);
  bb = wave_sum32(bb);
  const float ra = 1.0f / sqrtf(aa);   // precise div+sqrt (no fast-math): matches jnp
  const float rb = 1.0f / sqrtf(bb);
  float dd = 0.f;
#pragma unroll
  for (int k = 0; k < CHUNKS; ++k) {
    float t;
    t = fmaf(av[k].x, ra, fmaf(-bv[k].x, rb, EPSF)); dd = fmaf(t, t, dd);
    t = fmaf(av[k].y, ra, fmaf(-bv[k].y, rb, EPSF)); dd = fmaf(t, t, dd);
    t = fmaf(av[k].z, ra, fmaf(-bv[k].z, rb, EPSF)); dd = fmaf(t, t, dd);
    t = fmaf(av[k].w, ra, fmaf(-bv[k].w, rb, EPSF)); dd = fmaf(t, t, dd);
  }
  dd = wave_sum32(dd);
  const float value = 1.0f - sqrtf(dd);
  return 1.0f / (1.0f + expf(-value));
}

__global__ __launch_bounds__(BLOCK_THREADS) void
euclid_hash_decoder_kernel(const float* __restrict__ z,
                           const long long* __restrict__ src,
                           const long long* __restrict__ dst,
                           float* __restrict__ out,
                           int n_edges) {
  const int lane   = threadIdx.x & 31;
  const int wib    = threadIdx.x >> 5;            // wave index in block
  const int wid    = blockIdx.x * WPB + wib;      // global wave id == first edge
  const int stride = gridDim.x * WPB;
  const int off0   = lane * 4;                    // this lane's float offset in a 128-float chunk
  if (wid >= n_edges) return;                     // wave-uniform

#if USE_ASYNC
  // Double-buffered per-wave staging: buf = [rowA(512f) | rowB(512f)] = 4KB; x2 buffers.
  __shared__ __attribute__((aligned(16))) float smem[WPB][2][2 * D_FEAT];
  float* const bufs[2] = { &smem[wib][0][0], &smem[wib][1][0] };

  auto stage = [&](int e, float* buf) {
    const long long s = src[e];
    const long long t = dst[e];
    const float* rowA = z + (size_t)s * D_FEAT;
    const float* rowB = z + (size_t)t * D_FEAT;
#pragma unroll
    for (int k = 0; k < CHUNKS; ++k) {
      const int off = k * 128 + off0;             // 32 lanes x 16B = 512B per op
      __builtin_amdgcn_global_load_async_to_lds_b128(
          (gv4i_t*)(rowA + off), (lv4i_t*)(buf + off), 0, 0);
      __builtin_amdgcn_global_load_async_to_lds_b128(
          (gv4i_t*)(rowB + off), (lv4i_t*)(buf + D_FEAT + off), 0, 0);
    }
  };

  stage(wid, bufs[0]);                            // prime the pipeline (8 async ops)
  int cur = 0;
  for (int e = wid; e < n_edges; e += stride) {
    const int en = e + stride;
    const bool more = (en < n_edges);
    if (more) stage(en, bufs[cur ^ 1]);           // overlap next gather with this compute
    if (more) { WAIT_ASYNC(8); } else { WAIT_ASYNC(0); }  // in-order: current buf complete

    const float* buf = bufs[cur];
    float4 av[CHUNKS], bv[CHUNKS];
#pragma unroll
    for (int k = 0; k < CHUNKS; ++k) {            // conflict-free ds_load_b128
      av[k] = *(const float4*)(buf + k * 128 + off0);
      bv[k] = *(const float4*)(buf + D_FEAT + k * 128 + off0);
    }
    const float val = edge_value(av, bv);
    if (lane == 0) out[e] = val;
    cur ^= 1;
  }
#else
  // Fallback: direct coalesced b128 gathers + global_prefetch_b8 of next edge's rows.
  for (int e = wid; e < n_edges; e += stride) {
    const long long s = src[e];
    const long long t = dst[e];
    const float* rowA = z + (size_t)s * D_FEAT;
    const float* rowB = z + (size_t)t * D_FEAT;

    const int en = e + stride;
    if (en < n_edges) {
      const long long sn = src[en];
      const long long tn = dst[en];
      // 2KB row = 16 x 128B lines; lanes 0..15 cover next rowA, 16..31 next rowB.
      const char* pn = (lane < 16)
          ? (const char*)(z + (size_t)sn * D_FEAT) + lane * 128
          : (const char*)(z + (size_t)tn * D_FEAT) + (lane - 16) * 128;
      __builtin_prefetch(pn, 0, 3);               // -> global_prefetch_b8
    }

    float4 av[CHUNKS], bv[CHUNKS];
#pragma unroll
    for (int k = 0; k < CHUNKS; ++k) {            // global_load_b128, fully coalesced
      av[k] = *(const float4*)(rowA + k * 128 + off0);
      bv[k] = *(const float4*)(rowB + k * 128 + off0);
    }
    const float val = edge_value(av, bv);
    if (lane == 0) out[e] = val;
  }
#endif
}

extern "C" void kernel_launch(void* const* d_in, const int* in_sizes, int n_in,
                              void* d_out, int out_size, void* d_ws, size_t ws_size,
                              hipStream_t stream) {
  (void)n_in; (void)out_size; (void)d_ws; (void)ws_size;
  const float* z        = (const float*)d_in[0];
  const long long* ei   = (const long long*)d_in[1];   // int64 edge_index, [2, n_edges]
  const int n_edges     = in_sizes[1] / 2;
  const long long* src  = ei;
  const long long* dst  = ei + n_edges;
  float* out            = (float*)d_out;

  int blocks = (n_edges + WPB - 1) / WPB;
  if (blocks > MAX_BLOCKS) blocks = MAX_BLOCKS;
  if (blocks < 1) blocks = 1;
  euclid_hash_decoder_kernel<<<blocks, BLOCK_THREADS, 0, stream>>>(z, src, dst, out, n_edges);
}